// myConv2d_2611340116293
// MI455X (gfx1250) — compile-verified
//
#include <hip/hip_runtime.h>

typedef float v2f __attribute__((ext_vector_type(2)));
typedef float v8f __attribute__((ext_vector_type(8)));

#define BATCH 32
#define CIN   3
#define HIN   512
#define WIN   512
#define COUT  16
#define HO    510
#define WO    510
#define KTOT  27      // 3 channels * 3x3 taps
#define KCH   7       // ceil(27/4) K-chunks of 4
#define TILES_W 32    // ceil(510/16)
#define ROWS  6       // output rows per wave (510 = 85 * 6)
#define ROWGRP 85
#define WPB   8       // waves per block == tiles per block
#define TGRPS (TILES_W / WPB)   // 4 tile groups
#define LSTR  18      // padded LDS row stride (words); even -> 8B-aligned b64

__global__ __launch_bounds__(256)
void conv2d_wmma_f32(const float* __restrict__ x,
                     const float* __restrict__ wt,     // (16,3,3,3) = [n][k], k=c*9+kh*3+kw
                     const float* __restrict__ bias,   // (1,16)
                     float* __restrict__ out)          // (32,16,510,510)
{
    const int lane    = threadIdx.x & 31;
    const int waveBlk = threadIdx.x >> 5;

    // ---- block-uniform decomposition: b / row-group live in SGPRs ----
    const int bx  = blockIdx.x;
    const int tg  = bx & (TGRPS - 1);
    int tmp       = bx >> 2;
    const int ohg = tmp % ROWGRP;
    const int b   = tmp / ROWGRP;
    const int oh0 = ohg * ROWS;

    const int tile = tg * WPB + waveBlk;   // wave-level: only affects VGPR offsets
    const int ow0  = tile * 16;

    const int half = lane >> 4;   // K-half selector for A/B fragments
    const int mn   = lane & 15;   // M (pixel) for A / N (out-channel) for B/C

    // uniform (scalar) base pointers
    const float* xb = x   + ((size_t)b * CIN * HIN * WIN + (size_t)oh0 * WIN);
    float*       ob = out + ((size_t)(b * COUT) * HO + oh0) * (size_t)WO;

    // ---------------- B fragments: weights, zero-padded past K=26 ----------------
    v2f bf[KCH];
#pragma unroll
    for (int q = 0; q < KCH; ++q) {
        const int k0 = 4 * q + 2 * half;
        bf[q].x = (k0     < KTOT) ? wt[mn * KTOT + k0    ] : 0.0f;
        bf[q].y = (k0 + 1 < KTOT) ? wt[mn * KTOT + k0 + 1] : 0.0f;
    }

    // ---------------- A gather offsets: 32-bit, relative to scalar base ----------
    int iw = ow0 + mn;                 // this lane's pixel column
    if (iw > WO - 1) iw = WO - 1;      // clamp edge-tile loads (stores masked below)
    int aoff[KCH][2];
#pragma unroll
    for (int q = 0; q < KCH; ++q) {
#pragma unroll
        for (int j = 0; j < 2; ++j) {
            int k = 4 * q + 2 * half + j;
            if (k > KTOT - 1) k = KTOT - 1;      // clamped tail; B is 0 there
            const int c  = k / 9;
            const int r  = k % 9;
            const int kh = r / 3;
            const int kw = r % 3;
            aoff[q][j] = (c * HIN + kh) * WIN + kw + iw;
        }
    }

    const float bv = bias[mn];

    // store-side lane mapping: 4 b64 stores cover 16 channels x 16 pixels
    const int nq     = lane >> 3;        // channel subgroup 0..3
    const int mstart = 2 * (lane & 7);   // pixel-pair start
    const bool valid = (ow0 + mstart) < WO;   // edge tile: pair {14,15} masked

    __shared__ float lds[WPB][16 * LSTR];
    float* tl = &lds[waveBlk][0];

#pragma unroll
    for (int row = 0; row < ROWS; ++row) {
        // ---------------- C init = bias (fused) ----------------
        v8f c;
#pragma unroll
        for (int r = 0; r < 8; ++r) c[r] = bv;

        // ---------------- K loop: 7 x v_wmma_f32_16x16x4_f32 ----------------
#pragma unroll
        for (int q = 0; q < KCH; ++q) {
            v2f a;
            a.x = xb[aoff[q][0] + row * WIN];   // saddr + v_off + imm(row*2048)
            a.y = xb[aoff[q][1] + row * WIN];
            c = __builtin_amdgcn_wmma_f32_16x16x4_f32(
                    /*neg_a=*/false, a, /*neg_b=*/false, bf[q],
                    /*c_mod=*/(short)0, c, /*reuse_a=*/false, /*reuse_b=*/false);
        }

        // ---------------- transpose to [n][m] in LDS (b64 writes) ----------------
        // c[r] = element(m = r + 8*half, n = mn); pairs (r, r+1) are consecutive m
#pragma unroll
        for (int r = 0; r < 8; r += 2) {
            float2 p; p.x = c[r]; p.y = c[r + 1];
            *(float2*)&tl[mn * LSTR + 8 * half + r] = p;
        }
        // same-wave LDS is in-order; make the write->read ordering explicit
        asm volatile("s_wait_dscnt 0" ::: "memory");
        __builtin_amdgcn_sched_barrier(0);

        // ---------------- coalesced b64 stores: 64B per channel row --------------
#pragma unroll
        for (int t = 0; t < 4; ++t) {
            const int n = 4 * t + nq;
            float2 p = *(const float2*)&tl[n * LSTR + mstart];
            if (valid)
                *(float2*)(ob + (size_t)n * HO * WO + ow0 + mstart + row * WO) = p;
        }
    }
}

extern "C" void kernel_launch(void* const* d_in, const int* in_sizes, int n_in,
                              void* d_out, int out_size, void* d_ws, size_t ws_size,
                              hipStream_t stream) {
    (void)in_sizes; (void)n_in; (void)out_size; (void)d_ws; (void)ws_size;
    const float* x    = (const float*)d_in[0];
    const float* wt   = (const float*)d_in[1];
    const float* bias = (const float*)d_in[2];
    float* out        = (float*)d_out;

    const int blocks = TGRPS * ROWGRP * BATCH;   // 4 * 85 * 32 = 10,880
    conv2d_wmma_f32<<<blocks, 256, 0, stream>>>(x, wt, bias, out);
}